// Model_11888469475851
// MI455X (gfx1250) — compile-verified
//
#include <hip/hip_runtime.h>
#include <hip/hip_bf16.h>
#include <math.h>

// Problem dimensions (fixed by the reference)
#define NB   32
#define CH   64
#define HH   224
#define WW   224
#define HP   228
#define WP   228
#define PLANE   (HH * WW)       // 50176
#define PPLANE  (HP * WP)       // 51984
#define NGROUPS ((NB * CH * HP * WP) / 4)   // 26,615,808 (=103968*256)

static __device__ __forceinline__ float area_inv() { return 1.0f / (float)PPLANE; }

typedef __attribute__((ext_vector_type(2))) float v2f;
typedef __attribute__((ext_vector_type(4))) float v4f;
typedef __attribute__((ext_vector_type(8))) float v8f;

// sigmoid via v_exp_f32 + v_rcp_f32 (cheap transcendentals; ~1e-6 rel err)
static __device__ __forceinline__ float fast_sigmoid(float x) {
    return __builtin_amdgcn_rcpf(1.0f + __expf(-x));
}

// Abramowitz-Stegun 7.1.26 erf, |err| < 1.5e-7 absolute. One exp, one rcp.
static __device__ __forceinline__ float fast_erf(float x) {
    float ax = fabsf(x);
    float t  = __builtin_amdgcn_rcpf(fmaf(0.3275911f, ax, 1.0f));
    float p  = t * fmaf(t, fmaf(t, fmaf(t, fmaf(t, 1.061405429f, -1.453152027f),
                                        1.421413741f), -0.284496736f), 0.254829592f);
    float r  = 1.0f - p * __expf(-ax * ax);
    return copysignf(r, x);
}

static __device__ __forceinline__ float gelu_exact(float v) {
    return 0.5f * v * (1.0f + fast_erf(v * 0.70710678118654752f));
}

// reflect-pad index map: padded index -> source index (PAD=2)
static __device__ __forceinline__ int reflect2(int p) {
    int s = p - 2;
    s = (s < 0) ? -s : s;
    s = (s >= 224) ? (446 - s) : s;
    return s;
}

// ---------------------------------------------------------------------------
// Pass 1: per-(n,c) weighted sums over the SOURCE plane.
// Reflection pad folded into multiplicity weights: rows/cols {1,2,221,222} x2.
//   psum[n*64+c] = sum over padded plane of gated = x_c * sigmoid(x_{c-1})
//   xsum[n*64+c] = sum over padded plane of pad_x
// One block per plane -> plain stores, no atomics, no ws init needed.
// ---------------------------------------------------------------------------
__global__ void __launch_bounds__(256) k_reduce(const float* __restrict__ x,
                                                float* __restrict__ psum,
                                                float* __restrict__ xsum) {
    const int plane = blockIdx.x;          // n*64 + c
    const int c    = plane & 63;
    const int n    = plane >> 6;
    const int cm1  = (c + 63) & 63;
    const float* __restrict__ xc = x + (size_t)plane * PLANE;
    const float* __restrict__ xg = x + ((size_t)(n * CH + cm1)) * PLANE;

    float sg = 0.0f, sx = 0.0f;
    for (int i = threadIdx.x; i < PLANE; i += 256) {
        int h = i / WW;
        int w = i - h * WW;
        float wh = (((unsigned)(h - 1) <= 1u) | ((unsigned)(h - 221) <= 1u)) ? 2.0f : 1.0f;
        float wv = (((unsigned)(w - 1) <= 1u) | ((unsigned)(w - 221) <= 1u)) ? 2.0f : 1.0f;
        float wt = wh * wv;
        float a = xc[i];
        float b = xg[i];                   // gate plane: L2 hit from neighbor block
        float g = a * fast_sigmoid(b);
        sg = fmaf(wt, g, sg);
        sx = fmaf(wt, a, sx);
    }
    // wave32 reduction, then cross-wave via LDS
    #pragma unroll
    for (int off = 16; off > 0; off >>= 1) {
        sg += __shfl_down(sg, off, 32);
        sx += __shfl_down(sx, off, 32);
    }
    __shared__ float lsg[8], lsx[8];
    int lane = threadIdx.x & 31, wv = threadIdx.x >> 5;
    if (lane == 0) { lsg[wv] = sg; lsx[wv] = sx; }
    __syncthreads();
    if (threadIdx.x == 0) {
        float tg = 0.0f, tx = 0.0f;
        #pragma unroll
        for (int i = 0; i < 8; ++i) { tg += lsg[i]; tx += lsx[i]; }
        psum[plane] = tg;
        xsum[plane] = tx;
    }
}

// ---------------------------------------------------------------------------
// Pass 2: BN batch statistics via V_WMMA_F32_16X16X4_F32.
// colsum_c = sum_n pooled[n][c] as ones(16x4) x B(4x16) chained over 8 K-chunks.
// A = all-ones makes the result invariant to K-slot permutation; N mapping
// follows the documented C/D striping (N = lane & 15). 128 threads = 4 waves,
// wave w owns channels [16w, 16w+16). Also emits scale = sigmoid(BN(pooled))
// and chmean = xsum / paddedArea.
// ---------------------------------------------------------------------------
__global__ void __launch_bounds__(128) k_stats(const float* __restrict__ psum,
                                               const float* __restrict__ xsum,
                                               const float* __restrict__ bnw,
                                               const float* __restrict__ bnb,
                                               float* __restrict__ scale,
                                               float* __restrict__ chmean) {
    const int tid  = threadIdx.x;
    const int lane = tid & 31;
    const int wv   = tid >> 5;
    const int c    = wv * 16 + (lane & 15);
    const int rb   = (lane < 16) ? 0 : 2;   // K rows held by this lane-half
    const float ai = area_inv();

    v2f ones = {1.0f, 1.0f};
    v8f acc  = {};
    v8f acc2 = {};
    #pragma unroll
    for (int k = 0; k < 8; ++k) {
        int n0 = 4 * k + rb;
        float p0 = psum[n0 * CH + c] * ai;
        float p1 = psum[(n0 + 1) * CH + c] * ai;
        v2f b  = {p0, p1};
        v2f b2 = {p0 * p0, p1 * p1};
        acc  = __builtin_amdgcn_wmma_f32_16x16x4_f32(false, ones, false, b,
                                                     (short)0, acc,  false, false);
        acc2 = __builtin_amdgcn_wmma_f32_16x16x4_f32(false, ones, false, b2,
                                                     (short)0, acc2, false, false);
    }
    // every D row equals the column sum for channel c (A was all-ones)
    float mu  = acc[0]  * (1.0f / (float)NB);
    float var = acc2[0] * (1.0f / (float)NB) - mu * mu;   // biased, matches BN
    float invstd = rsqrtf(var + 1e-5f);

    if (lane < 16) {   // lanes 16-31 hold duplicates; avoid double stores
        float w = bnw[c], bsh = bnb[c];
        for (int nn = 0; nn < NB; ++nn) {
            float pooled = psum[nn * CH + c] * ai;
            float normed = fmaf((pooled - mu) * invstd, w, bsh);
            scale[nn * CH + c] = fast_sigmoid(normed);
        }
    }
    for (int i = tid; i < NB * CH; i += 128) chmean[i] = xsum[i] * ai;
}

// ---------------------------------------------------------------------------
// Pass 3: fused pad + GLU + reweight + residual + GELU, streamed.
// One thread per 4 consecutive padded-W elements (228 = 57*4), 16B-aligned
// nontemporal B128 stores so the 426MB output stream doesn't pollute L2
// (L2 is needed for the c / c-1 double read of x).
// ---------------------------------------------------------------------------
__global__ void __launch_bounds__(256) k_out(const float* __restrict__ x,
                                             const float* __restrict__ scale,
                                             const float* __restrict__ chmean,
                                             const float* __restrict__ ps,
                                             float* __restrict__ out) {
    int gid = blockIdx.x * 256 + threadIdx.x;
    if (gid >= NGROUPS) return;
    int g   = gid % 57;          // 4-wide group within padded row
    int row = gid / 57;          // (n*64+c)*228 + ph
    int ph  = row % HP;
    int pc  = row / HP;          // n*64 + c
    int c   = pc & 63;
    int n   = pc >> 6;
    int cm1 = (c + 63) & 63;

    int h = reflect2(ph);
    const float* __restrict__ xc = x + (size_t)pc * PLANE + (size_t)h * WW;
    const float* __restrict__ xg = x + ((size_t)(n * CH + cm1)) * PLANE + (size_t)h * WW;

    float sc = scale[pc];
    float cm = chmean[pc];
    float pscale = ps[0];

    int pw0 = g * 4;
    v4f o;
    #pragma unroll
    for (int j = 0; j < 4; ++j) {
        int w = reflect2(pw0 + j);
        float a = xc[w];
        float b = xg[w];
        float gg = a * fast_sigmoid(b);
        float v  = fmaf(gg, sc, cm) * pscale;
        o[j] = gelu_exact(v);
    }
    size_t oidx = (size_t)row * WP + pw0;          // 16B aligned (912B rows)
    __builtin_nontemporal_store(o, (v4f*)(out + oidx));
}

// ---------------------------------------------------------------------------
extern "C" void kernel_launch(void* const* d_in, const int* in_sizes, int n_in,
                              void* d_out, int out_size, void* d_ws, size_t ws_size,
                              hipStream_t stream) {
    const float* x      = (const float*)d_in[0];
    const float* bnw    = (const float*)d_in[1];
    const float* bnb    = (const float*)d_in[2];
    const float* pscale = (const float*)d_in[3];
    float* out = (float*)d_out;

    float* psum   = (float*)d_ws;          // 2048 floats
    float* xsum   = psum + NB * CH;        // 2048
    float* scl    = psum + 2 * NB * CH;    // 2048
    float* cmean  = psum + 3 * NB * CH;    // 2048

    k_reduce<<<NB * CH, 256, 0, stream>>>(x, psum, xsum);
    k_stats <<<1, 128, 0, stream>>>(psum, xsum, bnw, bnb, scl, cmean);
    k_out   <<<NGROUPS / 256, 256, 0, stream>>>(x, scl, cmean, pscale, out);
}